// GAMC_20194936226511
// MI455X (gfx1250) — compile-verified
//
#include <hip/hip_runtime.h>

// ---------------------------------------------------------------------------
// GIN graph autoencoder loss, full-fp32, WMMA f32_16x16x4 GEMMs on gfx1250.
// Pooling over `batch` in the reference is dead code (outputs unused) -> skipped.
// ---------------------------------------------------------------------------

typedef float v2f __attribute__((ext_vector_type(2)));
typedef float v8f __attribute__((ext_vector_type(8)));

#define DIN   128
#define DOUT  256
#define DHID  512
#define SBLK  128   // stat partial blocks
#define LBLK  240   // loss partial blocks
#define EPS   1e-5f

static inline int grd(int n) { return (n + 255) / 256; }

// dst[i,f] = mask[i] ? 0 : x[i,f]
__global__ void k_mask_copy(float* __restrict__ dst, const float* __restrict__ x,
                            const int* __restrict__ mask, int total, int F) {
    int i = blockIdx.x * blockDim.x + threadIdx.x;
    if (i >= total) return;
    dst[i] = mask[i / F] ? 0.0f : x[i];
}

__global__ void k_copy(float* __restrict__ dst, const float* __restrict__ src, int n) {
    int i = blockIdx.x * blockDim.x + threadIdx.x;
    if (i < n) dst[i] = src[i];
}

// dst[si[e], f] += (mask && mask[gi[e]]) ? 0 : src[gi[e], f]
__global__ void k_scatter_add(float* __restrict__ dst, const float* __restrict__ src,
                              const int* __restrict__ gi, const int* __restrict__ si,
                              const int* __restrict__ mask, int E, int F) {
    int i = blockIdx.x * blockDim.x + threadIdx.x;
    if (i >= E * F) return;
    int e = i / F, f = i - e * F;
    int s = gi[e];
    if (mask && mask[s]) return;
    atomicAdd(&dst[si[e] * F + f], src[s * F + f]);
}

// C[N x M] = op(A)[N x K] @ W[K x M], op = optional per-column BN+ReLU on A.
// One block = 8 wave32 waves = one 16-row tile, all M columns.
template<int K, int M, bool BNRELU>
__global__ __launch_bounds__(256) void k_gemm(const float* __restrict__ A,
                                              const float* __restrict__ W,
                                              float* __restrict__ C,
                                              const float* __restrict__ g,
                                              const float* __restrict__ b,
                                              const float* __restrict__ mu,
                                              const float* __restrict__ rstd) {
    constexpr int TPW  = M / 128;   // 16-col tiles per wave
    constexpr int KC   = 64;        // K chunk staged in LDS
    constexpr int LSTR = KC + 4;    // pad: bank-conflict-free, 8B-aligned pairs
    __shared__ float lds[16 * LSTR];

    const int tid  = threadIdx.x;
    const int wave = tid >> 5;
    const int lane = tid & 31;
    const int m    = lane & 15;     // A row within tile / B,C column within tile
    const int kh   = lane >> 4;     // K half (f32 16x16x4 layout)
    const int rowBase = blockIdx.x * 16;

    v8f acc[TPW];
#pragma unroll
    for (int t = 0; t < TPW; ++t) acc[t] = (v8f)0.0f;

    for (int k0 = 0; k0 < K; k0 += KC) {
        // stage 16 x KC A-chunk, coalesced, with fused BN+ReLU
#pragma unroll
        for (int j = 0; j < (16 * KC) / 256; ++j) {
            int idx = tid + j * 256;
            int r = idx >> 6, c = idx & (KC - 1);
            float v = A[(rowBase + r) * K + k0 + c];
            if constexpr (BNRELU) {
                int f = k0 + c;
                v = g[f] * (v - mu[f]) * rstd[f] + b[f];
                v = v > 0.0f ? v : 0.0f;
            }
            lds[r * LSTR + c] = v;
        }
        __syncthreads();
#pragma unroll
        for (int t = 0; t < TPW; ++t) {
            const int col = (wave * TPW + t) * 16 + m;
#pragma unroll
            for (int kk = 0; kk < KC; kk += 4) {
                // A 16x4 f32 frag: lanes 0-15 -> K={kk,kk+1}, lanes 16-31 -> K={kk+2,kk+3}
                v2f a = *(const v2f*)&lds[m * LSTR + kk + 2 * kh];
                int kb = k0 + kk + 2 * kh;
                v2f bb;
                bb.x = W[kb * M + col];
                bb.y = W[(kb + 1) * M + col];
                acc[t] = __builtin_amdgcn_wmma_f32_16x16x4_f32(
                    false, a, false, bb, (short)0, acc[t], false, false);
            }
        }
        __syncthreads();
    }
    // C/D layout: VGPR r, lanes 0-15 -> tile row r; lanes 16-31 -> row r+8
#pragma unroll
    for (int t = 0; t < TPW; ++t) {
        const int col = (wave * TPW + t) * 16 + m;
#pragma unroll
        for (int r = 0; r < 8; ++r)
            C[(rowBase + r + kh * 8) * M + col] = acc[t][r];
    }
}

// per-column partial sums / sums-of-squares over a row range
__global__ void k_bn_stats_partial(const float* __restrict__ T, int n, int F,
                                   float* __restrict__ psum, float* __restrict__ psq) {
    int blk = blockIdx.x;
    int rpb = (n + gridDim.x - 1) / gridDim.x;
    int r0 = blk * rpb;
    int r1 = r0 + rpb; if (r1 > n) r1 = n;
    for (int c = threadIdx.x; c < F; c += blockDim.x) {
        float s = 0.0f, q = 0.0f;
        for (int r = r0; r < r1; ++r) {
            float v = T[r * F + c];
            s += v; q += v * v;
        }
        psum[blk * F + c] = s;
        psq [blk * F + c] = q;
    }
}

__global__ void k_bn_stats_final(const float* __restrict__ psum, const float* __restrict__ psq,
                                 int F, int nb, int n,
                                 float* __restrict__ mu, float* __restrict__ rstd) {
    for (int c = threadIdx.x; c < F; c += blockDim.x) {
        double s = 0.0, q = 0.0;
        for (int b = 0; b < nb; ++b) { s += psum[b * F + c]; q += psq[b * F + c]; }
        double mean = s / n;
        double var  = q / n - mean * mean;
        if (var < 0.0) var = 0.0;
        mu[c]   = (float)mean;
        rstd[c] = rsqrtf((float)var + EPS);
    }
}

// out = relu(g*(in-mu)*rstd + b), optionally zeroed where mask[node]
__global__ void k_bn_relu(float* __restrict__ out, const float* __restrict__ in,
                          const float* __restrict__ g, const float* __restrict__ b,
                          const float* __restrict__ mu, const float* __restrict__ rstd,
                          const int* __restrict__ mask, int total, int F) {
    int i = blockIdx.x * blockDim.x + threadIdx.x;
    if (i >= total) return;
    int node = i / F, f = i - node * F;
    float v = g[f] * (in[i] - mu[f]) * rstd[f] + b[f];
    v = v > 0.0f ? v : 0.0f;
    if (mask && mask[node]) v = 0.0f;
    out[i] = v;
}

// one wave per node: weighted sum of (1 - cos(A_i, B_i)); fixed-order reductions
__global__ void k_cos_partial(const float* __restrict__ A, const float* __restrict__ B,
                              const int* __restrict__ mask, int n, int F,
                              float* __restrict__ pnum, float* __restrict__ pden) {
    __shared__ float sN[8], sD[8];
    int tid = threadIdx.x, lane = tid & 31, wib = tid >> 5;
    int gw = (blockIdx.x * blockDim.x + tid) >> 5;
    int nw = (gridDim.x * blockDim.x) >> 5;
    float lnum = 0.0f, lden = 0.0f;
    for (int i = gw; i < n; i += nw) {
        float dab = 0.0f, na = 0.0f, nb = 0.0f;
        for (int f = lane; f < F; f += 32) {
            float a = A[i * F + f], bb = B[i * F + f];
            dab += a * bb; na += a * a; nb += bb * bb;
        }
        for (int o = 16; o > 0; o >>= 1) {
            dab += __shfl_xor(dab, o, 32);
            na  += __shfl_xor(na,  o, 32);
            nb  += __shfl_xor(nb,  o, 32);
        }
        float w = mask ? (mask[i] ? 1.0f : 0.0f) : 1.0f;
        float c = dab / (fmaxf(sqrtf(na), 1e-12f) * fmaxf(sqrtf(nb), 1e-12f));
        lnum += w * (1.0f - c);
        lden += w;
    }
    if (lane == 0) { sN[wib] = lnum; sD[wib] = lden; }
    __syncthreads();
    if (tid == 0) {
        float s = 0.0f, d = 0.0f;
        for (int w = 0; w < 8; ++w) { s += sN[w]; d += sD[w]; }
        pnum[blockIdx.x] = s;
        pden[blockIdx.x] = d;
    }
}

__global__ void k_reduce2(const float* __restrict__ pnum, const float* __restrict__ pden,
                          int nb, float* __restrict__ acc2) {
    if (blockIdx.x == 0 && threadIdx.x == 0) {
        float s = 0.0f, d = 0.0f;
        for (int i = 0; i < nb; ++i) { s += pnum[i]; d += pden[i]; }
        acc2[0] = s; acc2[1] = d;
    }
}

__global__ void k_final(float* __restrict__ out, const float* __restrict__ acc) {
    if (blockIdx.x == 0 && threadIdx.x == 0)
        out[0] = acc[0] / acc[1] + acc[2] / acc[3] + 0.1f * (acc[4] / acc[5]);
}

extern "C" void kernel_launch(void* const* d_in, const int* in_sizes, int n_in,
                              void* d_out, int out_size, void* d_ws, size_t ws_size,
                              hipStream_t stream) {
    (void)n_in; (void)out_size; (void)ws_size;
    const float* x     = (const float*)d_in[0];
    const int*   ei1   = (const int*)d_in[1];
    const int*   ei2   = (const int*)d_in[2];
    const int*   mask1 = (const int*)d_in[3];
    const int*   mask2 = (const int*)d_in[4];
    // d_in[5] (batch) intentionally unused: pooled encoder outputs are dead code.
    const float* e0w1 = (const float*)d_in[6];
    const float* e0w2 = (const float*)d_in[7];
    const float* e0bg = (const float*)d_in[8];
    const float* e0bb = (const float*)d_in[9];
    const float* e0ng = (const float*)d_in[10];
    const float* e0nb = (const float*)d_in[11];
    const float* e1w1 = (const float*)d_in[12];
    const float* e1w2 = (const float*)d_in[13];
    const float* e1bg = (const float*)d_in[14];
    const float* e1bb = (const float*)d_in[15];
    const float* e1ng = (const float*)d_in[16];
    const float* e1nb = (const float*)d_in[17];
    const float* dw1  = (const float*)d_in[18];
    const float* dw2  = (const float*)d_in[19];
    const float* dbg  = (const float*)d_in[20];
    const float* dbb  = (const float*)d_in[21];
    const float* dng  = (const float*)d_in[22];
    const float* dnb  = (const float*)d_in[23];

    const int N = in_sizes[0] / DIN;   // 50000 (multiple of 16)
    const int E = in_sizes[1] / 2;     // 320000

    // workspace layout (floats)
    float* B0   = (float*)d_ws;            // N x 256
    float* B1   = B0 + (size_t)N * DOUT;   // N x 512
    float* B2   = B1 + (size_t)N * DHID;   // N x 512
    float* R1   = B2 + (size_t)N * DHID;   // N x 128
    float* R2   = R1 + (size_t)N * DIN;    // N x 128
    float* psum = R2 + (size_t)N * DIN;    // SBLK x 512
    float* psq  = psum + SBLK * DHID;      // SBLK x 512
    float* mu   = psq  + SBLK * DHID;      // 512
    float* rstd = mu + DHID;               // 512
    float* pnum = rstd + DHID;             // LBLK
    float* pden = pnum + LBLK;             // LBLK
    float* acc  = pden + LBLK;             // 6

    auto stats = [&](const float* T, int F) {
        k_bn_stats_partial<<<SBLK, 256, 0, stream>>>(T, N, F, psum, psq);
        k_bn_stats_final<<<1, 256, 0, stream>>>(psum, psq, F, SBLK, N, mu, rstd);
    };

    auto run_pass = [&](const int* ei, const int* mask, float* R) {
        const int* gi = ei;        // ei[0]: gather
        const int* si = ei + E;    // ei[1]: scatter
        // ---- encoder layer 0 (IN=128 -> HID -> OUT=256)
        k_mask_copy<<<grd(N * DIN), 256, 0, stream>>>(B0, x, mask, N * DIN, DIN);
        k_scatter_add<<<grd(E * DIN), 256, 0, stream>>>(B0, x, gi, si, mask, E, DIN);
        k_gemm<DIN, DHID, false><<<N / 16, 256, 0, stream>>>(B0, e0w1, B1, nullptr, nullptr, nullptr, nullptr);
        stats(B1, DHID);
        k_gemm<DHID, DOUT, true><<<N / 16, 256, 0, stream>>>(B1, e0w2, B2, e0bg, e0bb, mu, rstd);
        stats(B2, DOUT);
        k_bn_relu<<<grd(N * DOUT), 256, 0, stream>>>(B0, B2, e0ng, e0nb, mu, rstd, nullptr, N * DOUT, DOUT);
        // ---- encoder layer 1 (256 -> HID -> 256)
        k_copy<<<grd(N * DOUT), 256, 0, stream>>>(B1, B0, N * DOUT);
        k_scatter_add<<<grd(E * DOUT), 256, 0, stream>>>(B1, B0, gi, si, nullptr, E, DOUT);
        k_gemm<DOUT, DHID, false><<<N / 16, 256, 0, stream>>>(B1, e1w1, B2, nullptr, nullptr, nullptr, nullptr);
        stats(B2, DHID);
        k_gemm<DHID, DOUT, true><<<N / 16, 256, 0, stream>>>(B2, e1w2, B0, e1bg, e1bb, mu, rstd);
        stats(B0, DOUT);
        // re_h = mask(relu(bn(u1)))
        k_bn_relu<<<grd(N * DOUT), 256, 0, stream>>>(B1, B0, e1ng, e1nb, mu, rstd, mask, N * DOUT, DOUT);
        // ---- decoder (256 -> HID -> 128)
        k_copy<<<grd(N * DOUT), 256, 0, stream>>>(B2, B1, N * DOUT);
        k_scatter_add<<<grd(E * DOUT), 256, 0, stream>>>(B2, B1, gi, si, nullptr, E, DOUT);
        k_gemm<DOUT, DHID, false><<<N / 16, 256, 0, stream>>>(B2, dw1, B1, nullptr, nullptr, nullptr, nullptr);
        stats(B1, DHID);
        k_gemm<DHID, DIN, true><<<N / 16, 256, 0, stream>>>(B1, dw2, B0, dbg, dbb, mu, rstd);
        stats(B0, DIN);
        k_bn_relu<<<grd(N * DIN), 256, 0, stream>>>(R, B0, dng, dnb, mu, rstd, nullptr, N * DIN, DIN);
    };

    run_pass(ei1, mask1, R1);
    k_cos_partial<<<LBLK, 256, 0, stream>>>(R1, x, mask1, N, DIN, pnum, pden);
    k_reduce2<<<1, 1, 0, stream>>>(pnum, pden, LBLK, acc + 0);

    run_pass(ei2, mask2, R2);
    k_cos_partial<<<LBLK, 256, 0, stream>>>(R2, x, mask2, N, DIN, pnum, pden);
    k_reduce2<<<1, 1, 0, stream>>>(pnum, pden, LBLK, acc + 2);

    k_cos_partial<<<LBLK, 256, 0, stream>>>(R2, R1, nullptr, N, DIN, pnum, pden);
    k_reduce2<<<1, 1, 0, stream>>>(pnum, pden, LBLK, acc + 4);

    k_final<<<1, 1, 0, stream>>>((float*)d_out, acc);
}